// RandomSampler_66786741453008
// MI455X (gfx1250) — compile-verified
//
#include <hip/hip_runtime.h>
#include <hip/hip_bf16.h>
#include <stdint.h>

// Problem constants (match reference: B, TG, A, N_RULES, K)
#define B_   32
#define TG_  131072
#define A_   8
#define K_   16384
#define NB   4096            // buckets per select level
#define CH   256             // TDM tile: 256 f32 = 1 KB
#define PARTS 8              // histogram slices per row (blocks per row)

typedef unsigned int u32x4 __attribute__((ext_vector_type(4)));
typedef unsigned int u32x8 __attribute__((ext_vector_type(8)));

// ---------------------------------------------------------------------------
// ws layout (ints): part1[B][8][NB] | sfx1[B][NB] | sfx2[B][NB] |
//                   rank1[B][NB] | hist2[B][NB] | rank2[B][NB] | rowinfo[B][8]
// rowinfo: [0]=tb [1]=cAboveTb [2]=K2(=K-cAboveTb) [3]=ftb [4]=fbase [5]=rem2
// ---------------------------------------------------------------------------

__global__ void rs_zero_kernel(int* __restrict__ p, int n) {
    for (int i = blockIdx.x * blockDim.x + threadIdx.x; i < n;
         i += gridDim.x * blockDim.x)
        p[i] = 0;
}

// ---------------------------------------------------------------------------
// TDM 1-D tile descriptor issue: count=1, type=2(image), data_size=4B,
// tile_dim0 = tensor_dim0 = CH, tile_dim1 = tensor_dim1 = 1.
// Groups per CDNA5 ISA.
// ---------------------------------------------------------------------------
__device__ __forceinline__ void tdm_load_1d(const void* gptr, unsigned ldsOff) {
    unsigned long long ga = (unsigned long long)gptr;
    unsigned galo = (unsigned)__builtin_amdgcn_readfirstlane((int)(unsigned)ga);
    unsigned gahi = (unsigned)__builtin_amdgcn_readfirstlane((int)(unsigned)(ga >> 32));
    unsigned lo   = (unsigned)__builtin_amdgcn_readfirstlane((int)ldsOff);
    u32x4 g0 = { 1u,                                   // count=1, user D#
                 lo,                                   // lds_addr
                 galo,                                 // global_addr[31:0]
                 (gahi & 0x01FFFFFFu) | 0x80000000u }; // addr[56:32] | type=2
    u32x8 g1 = { 0x00020000u,                          // data_size = 4B
                 ((unsigned)CH & 0xFFFFu) << 16,       // tensor_dim0[15:0] @[63:48]
                 0x00010000u,                          // tensor_dim1 = 1  @[95:80]
                 ((unsigned)CH & 0xFFFFu) << 16,       // tile_dim0  @[127:112]
                 0x00000001u,                          // tile_dim1 = 1 @[143:128]
                 (unsigned)CH,                         // tensor_dim0_stride[31:0]
                 0u, 0u };
    asm volatile("tensor_load_to_lds %0, %1" :: "s"(g0), "s"(g1) : "memory");
}

// ---------------------------------------------------------------------------
// Level-1 histogram. 8 blocks/row, 8 waves/block; each wave streams 8 tiles
// of 256 f32 for rand & mask through its private LDS double buffer via the
// Tensor Data Mover (tensor_load_to_lds + s_wait_tensorcnt), histograms into
// shared LDS (ds_add), then the block stores its private histogram slice with
// global_store_async_from_lds_b128 (ASYNCcnt) -- no global atomics at all.
// ---------------------------------------------------------------------------
__global__ __launch_bounds__(256)
void rs_hist1_kernel(const float* __restrict__ rnd, const float* __restrict__ msk,
                     int* __restrict__ part) {
    __shared__ int   lhist[NB];
    __shared__ __align__(16) float lrand[8][2][CH];
    __shared__ __align__(16) float lmask[8][2][CH];

    const int t    = threadIdx.x;
    const int w    = t >> 5;            // wave id 0..7
    const int lane = t & 31;

    for (int i = t; i < NB; i += 256) lhist[i] = 0;
    __syncthreads();

    const int    row     = blockIdx.x / PARTS;
    const int    sub     = blockIdx.x % PARTS;
    const size_t rowBase = (size_t)row * TG_;
    // 512 tiles per row; this wave owns 8 consecutive tiles
    const int firstTile = sub * 64 + w * 8;

    auto issue = [&](int k) {
        const size_t elem = rowBase + (size_t)(firstTile + k) * CH;
        const int    buf  = k & 1;
        tdm_load_1d(rnd + elem, (unsigned)(size_t)(&lrand[w][buf][0]));
        tdm_load_1d(msk + elem, (unsigned)(size_t)(&lmask[w][buf][0]));
    };

    issue(0);
    for (int k = 0; k < 8; ++k) {
        if (k < 7) {
            issue(k + 1);
            // next tile's 2 TDM ops may remain outstanding; tile k is done
            // once this wave's TENSORcnt <= 2 (TDM is in-order per wave)
            asm volatile("s_wait_tensorcnt 0x2" ::: "memory");
        } else {
            asm volatile("s_wait_tensorcnt 0x0" ::: "memory");
        }
        const int buf = k & 1;
#pragma unroll
        for (int j = 0; j < CH / 32; ++j) {
            const int e = lane + 32 * j;
            float s = lrand[w][buf][e] * lmask[w][buf][e];
            int   q = (int)(s * 4096.0f);
            q = q > (NB - 1) ? (NB - 1) : q;
            atomicAdd(&lhist[q], 1);            // ds_add_u32
        }
    }
    __syncthreads();

    // async store of this block's private histogram slice (4096 ints)
    int* dst = part + ((size_t)(row * PARTS + sub)) * NB;
#pragma unroll
    for (int r = 0; r < 4; ++r) {
        const int idx = (r * 256 + t) * 4;
        int*     gp   = dst + idx;
        unsigned lsrc = (unsigned)(size_t)(&lhist[idx]);
        asm volatile("global_store_async_from_lds_b128 %0, %1, off"
                     :: "v"(gp), "v"(lsrc) : "memory");
    }
    asm volatile("s_wait_asynccnt 0x0" ::: "memory");
}

// ---------------------------------------------------------------------------
// Suffix-scan over NB buckets (descending) of nparts summed slices; find the
// threshold bucket. level 0: Kq = K_; level 1: Kq = rowinfo[2].
// ---------------------------------------------------------------------------
__global__ __launch_bounds__(256)
void rs_scan_kernel(const int* __restrict__ hist, int* __restrict__ sfx,
                    int* __restrict__ rowinfo, int level, int nparts) {
    __shared__ int psum[256];
    const int row = blockIdx.x;
    const int t   = threadIdx.x;
    const int* h  = hist + (size_t)row * nparts * NB;
    int*       sf = sfx  + (size_t)row * NB;
    int*       ri = rowinfo + row * 8;
    const int  Kq = (level == 0) ? K_ : ri[2];

    int lc[16];
    int s = 0;
#pragma unroll
    for (int j = 0; j < 16; ++j) {
        const int b = NB - 1 - (t * 16 + j);   // descending buckets
        int c = 0;
        for (int p = 0; p < nparts; ++p) c += h[p * NB + b];
        lc[j] = c;
        s += c;
    }
    psum[t] = s;
    __syncthreads();
    for (int off = 1; off < 256; off <<= 1) {   // Hillis-Steele inclusive scan
        int v = (t >= off) ? psum[t - off] : 0;
        __syncthreads();
        psum[t] += v;
        __syncthreads();
    }
    int run = (t == 0) ? 0 : psum[t - 1];       // count above my chunk
#pragma unroll
    for (int j = 0; j < 16; ++j) {
        const int b = NB - 1 - (t * 16 + j);
        sf[b] = run;                            // count strictly above bucket b
        const int c = lc[j];
        if (c > 0 && run <= Kq && Kq < run + c) {
            if (level == 0) { ri[0] = b; ri[1] = run; ri[2] = K_ - run; }
            else            { ri[3] = b; ri[4] = run; ri[5] = Kq - run; }
        }
        run += c;
    }
}

// ---------------------------------------------------------------------------
// Level-2 histogram over the threshold bucket only (key low 12 bits; exact
// because each coarse bucket q>=1 lies inside one binade).
// ---------------------------------------------------------------------------
__global__ __launch_bounds__(256)
void rs_hist2_kernel(const float* __restrict__ rnd, const float* __restrict__ msk,
                     const int* __restrict__ rowinfo, int* __restrict__ hist2) {
    const int gid     = blockIdx.x * 256 + threadIdx.x;
    const int per_row = TG_ / 4;
    const int row     = gid / per_row;
    const int e4      = gid % per_row;
    const int tb      = rowinfo[row * 8 + 0];

    const float4 r4 = ((const float4*)(rnd + (size_t)row * TG_))[e4];
    const float4 m4 = ((const float4*)(msk + (size_t)row * TG_))[e4];
    const float rs[4] = {r4.x, r4.y, r4.z, r4.w};
    const float ms[4] = {m4.x, m4.y, m4.z, m4.w};
    int* h2 = hist2 + (size_t)row * NB;
#pragma unroll
    for (int j = 0; j < 4; ++j) {
        float s = rs[j] * ms[j];
        int   q = (int)(s * 4096.0f);
        q = q > (NB - 1) ? (NB - 1) : q;
        if (q == tb) {
            const int f = (int)(__float_as_uint(s) & 0xFFFu);
            atomicAdd(&h2[f], 1);
        }
    }
}

// ---------------------------------------------------------------------------
// Compaction + gather. Output ordered by (coarse, fine) bucket descending.
// Prefetch the body row (global_prefetch_b8) while the slot atomic is in
// flight, then copy 32 B with two b128 transactions.
// ---------------------------------------------------------------------------
__global__ __launch_bounds__(256)
void rs_compact_kernel(const float* __restrict__ rnd, const float* __restrict__ msk,
                       const float* __restrict__ body, const long long* __restrict__ rule,
                       const int* __restrict__ rowinfo,
                       const int* __restrict__ sfx1, int* __restrict__ rank1,
                       const int* __restrict__ sfx2, int* __restrict__ rank2,
                       float* __restrict__ out_body, float* __restrict__ out_mask,
                       float* __restrict__ out_rule) {
    const int gid     = blockIdx.x * 256 + threadIdx.x;
    const int per_row = TG_ / 4;
    const int row     = gid / per_row;
    const int e4      = gid % per_row;

    const int* ri   = rowinfo + row * 8;
    const int  tb   = ri[0];
    const int  cAb  = ri[1];
    const int  ftb  = ri[3];
    const int  rem2 = ri[5];

    const float4 r4 = ((const float4*)(rnd + (size_t)row * TG_))[e4];
    const float4 m4 = ((const float4*)(msk + (size_t)row * TG_))[e4];
    const float rs[4] = {r4.x, r4.y, r4.z, r4.w};
    const float ms[4] = {m4.x, m4.y, m4.z, m4.w};

    const int* s1 = sfx1  + (size_t)row * NB;
    int*       k1 = rank1 + (size_t)row * NB;
    const int* s2 = sfx2  + (size_t)row * NB;
    int*       k2 = rank2 + (size_t)row * NB;

#pragma unroll
    for (int j = 0; j < 4; ++j) {
        const int i = e4 * 4 + j;
        float s = rs[j] * ms[j];
        int   q = (int)(s * 4096.0f);
        q = q > (NB - 1) ? (NB - 1) : q;
        if (q < tb) continue;

        const float* brow = body + ((size_t)row * TG_ + (size_t)i) * A_;
        __builtin_prefetch(brow, 0, 1);        // global_prefetch_b8

        int slot = -1;
        if (q > tb) {
            slot = s1[q] + atomicAdd(&k1[q], 1);
        } else {
            const int f = (int)(__float_as_uint(s) & 0xFFFu);
            if (f > ftb) {
                slot = cAb + s2[f] + atomicAdd(&k2[f], 1);
            } else if (f == ftb) {
                const int r = atomicAdd(&k2[f], 1);
                if (r < rem2) slot = cAb + s2[f] + r;
            }
        }
        if (slot >= 0 && slot < K_) {
            const float4 b0 = ((const float4*)brow)[0];
            const float4 b1 = ((const float4*)brow)[1];
            float4* orow = (float4*)(out_body + ((size_t)row * K_ + (size_t)slot) * A_);
            orow[0] = b0;
            orow[1] = b1;
            out_mask[(size_t)row * K_ + slot] = ms[j];
            out_rule[(size_t)row * K_ + slot] = (float)rule[(size_t)row * TG_ + i];
        }
    }
}

// ---------------------------------------------------------------------------
extern "C" void kernel_launch(void* const* d_in, const int* in_sizes, int n_in,
                              void* d_out, int out_size, void* d_ws, size_t ws_size,
                              hipStream_t stream) {
    const float*     body = (const float*)d_in[0];      // [B, TG, A] f32
    const float*     msk  = (const float*)d_in[1];      // [B, TG]    f32
    const long long* rule = (const long long*)d_in[2];  // [B, TG]    i64
    const float*     rnd  = (const float*)d_in[3];      // [B, TG]    f32
    // d_in[4] = k (compile-time K_)

    int* ws      = (int*)d_ws;
    int* part1   = ws;                                  // B*PARTS*NB (fully overwritten)
    int* sfx1    = part1 + (size_t)B_ * PARTS * NB;     // B*NB (fully overwritten)
    int* sfx2    = sfx1  + (size_t)B_ * NB;             // B*NB (fully overwritten)
    int* rank1   = sfx2  + (size_t)B_ * NB;             // B*NB (zeroed)
    int* hist2   = rank1 + (size_t)B_ * NB;             // B*NB (zeroed)
    int* rank2   = hist2 + (size_t)B_ * NB;             // B*NB (zeroed)
    int* rowinfo = rank2 + (size_t)B_ * NB;             // B*8  (zeroed)
    const int zero_ints = 3 * B_ * NB + B_ * 8;

    float* out_body = (float*)d_out;                       // B*K*A
    float* out_mask = out_body + (size_t)B_ * K_ * A_;     // B*K
    float* out_rule = out_mask + (size_t)B_ * K_;          // B*K

    rs_zero_kernel<<<512, 256, 0, stream>>>(rank1, zero_ints);
    rs_hist1_kernel<<<B_ * PARTS, 256, 0, stream>>>(rnd, msk, part1);
    rs_scan_kernel<<<B_, 256, 0, stream>>>(part1, sfx1, rowinfo, 0, PARTS);
    rs_hist2_kernel<<<(B_ * TG_ / 4) / 256, 256, 0, stream>>>(rnd, msk, rowinfo, hist2);
    rs_scan_kernel<<<B_, 256, 0, stream>>>(hist2, sfx2, rowinfo, 1, 1);
    rs_compact_kernel<<<(B_ * TG_ / 4) / 256, 256, 0, stream>>>(
        rnd, msk, body, rule, rowinfo, sfx1, rank1, sfx2, rank2,
        out_body, out_mask, out_rule);
}